// PatchToPatchEdgeConvolution_1580547965681
// MI455X (gfx1250) — compile-verified
//
#include <hip/hip_runtime.h>

// Problem constants from the reference
constexpr int N = 4096;    // nodes
constexpr int E = 32768;   // edges
constexpr int B = 256;     // batch (query edges)
constexpr int D = 256;     // edge feature dim
constexpr int K = 256;     // output dim

typedef __attribute__((ext_vector_type(2))) float v2f;
typedef __attribute__((ext_vector_type(4))) float v4f;
typedef __attribute__((ext_vector_type(8))) float v8f;

// ---------------------------------------------------------------------------
// Kernel 0: initialize atomic min/max arrays in workspace
// ---------------------------------------------------------------------------
__global__ void init_ws_kernel(int* __restrict__ eu, int* __restrict__ ev,
                               int* __restrict__ bmin, int* __restrict__ bmax) {
  int i = blockIdx.x * blockDim.x + threadIdx.x;
  if (i < E) { eu[i] = 0x7FFFFFFF; ev[i] = -1; }
  if (i < B) { bmin[i] = 0x7FFFFFFF; bmax[i] = -1; }
}

// ---------------------------------------------------------------------------
// Kernel 1: per-batch endpoints from edge_nodes [B,N] (exactly two 1s per row)
// ---------------------------------------------------------------------------
__global__ void batch_endpoints_kernel(const float* __restrict__ en,
                                       int* __restrict__ bmin, int* __restrict__ bmax) {
  int i = blockIdx.x * blockDim.x + threadIdx.x;   // B*N = 1,048,576 threads
  float v = en[i];
  if (v != 0.0f) {
    int b = i >> 12;        // / N
    int n = i & (N - 1);    // % N
    atomicMin(&bmin[b], n);
    atomicMax(&bmax[b], n);
  }
}

// ---------------------------------------------------------------------------
// Kernel 2: per-edge endpoints from inc [N,E] (exactly two 1s per column).
// 536 MB streaming scan -> non-temporal float4 loads + software prefetch.
// This is the memory-roofline-dominant stage (~23 us at 23.3 TB/s).
// ---------------------------------------------------------------------------
__global__ void edge_endpoints_kernel(const v4f* __restrict__ inc4,
                                      int* __restrict__ eu, int* __restrict__ ev) {
  const long total4 = (long)N * E / 4;             // 33,554,432 float4s
  const long stride = (long)gridDim.x * blockDim.x;
  for (long i = (long)blockIdx.x * blockDim.x + threadIdx.x; i < total4; i += stride) {
    if (i + stride < total4)
      __builtin_prefetch(&inc4[i + stride], 0, 1); // global_prefetch_b8
    v4f v = __builtin_nontemporal_load(&inc4[i]);  // NT: don't pollute L2
    long base = i * 4;
    int n  = (int)(base >> 15);                    // / E (row in [N,E])
    int e0 = (int)(base & (E - 1));                // % E
#pragma unroll
    for (int j = 0; j < 4; ++j) {
      if (v[j] != 0.0f) {
        atomicMin(&eu[e0 + j], n);
        atomicMax(&ev[e0 + j], n);
      }
    }
  }
}

// ---------------------------------------------------------------------------
// Kernel 3: h_edges = edge_feats @ weight + bias   [E,D]@[D,K] in f32 WMMA.
// One 16x16 output tile per wave32, K-dim stepped by 4 (V_WMMA_F32_16X16X4_F32).
// Per ISA 7.12.2 f32 layouts:
//   A 16x4 : lanes 0-15 hold K=kk,kk+1 ; lanes 16-31 hold K=kk+2,kk+3 (M = lane&15)
//   B 4x16 : VGPR r, lanes 0-15 -> K=kk+r ; lanes 16-31 -> K=kk+2+r  (N = lane&15)
//   C 16x16: VGPR i, lanes 0-15 -> M=i ; lanes 16-31 -> M=i+8
// ---------------------------------------------------------------------------
__global__ void __launch_bounds__(256)
gemm_hedges_kernel(const float* __restrict__ ef, const float* __restrict__ w,
                   const float* __restrict__ bias, float* __restrict__ h) {
  const int wave = threadIdx.x >> 5;
  const int lane = threadIdx.x & 31;
  const int tile = blockIdx.x * 8 + wave;          // 32768 tiles total
  const int m0 = (tile >> 4) * 16;                 // K/16 = 16 n-tiles
  const int n0 = (tile & 15) * 16;
  const int half = lane >> 4;
  const int lm   = lane & 15;

  v8f c = {};
  const float* arow = ef + (long)(m0 + lm) * D;
  for (int kk = 0; kk < D; kk += 4) {
    v2f a = *(const v2f*)(arow + kk + 2 * half);
    v2f b;
    b[0] = w[(kk + 2 * half + 0) * K + n0 + lm];
    b[1] = w[(kk + 2 * half + 1) * K + n0 + lm];
    c = __builtin_amdgcn_wmma_f32_16x16x4_f32(
        /*neg_a=*/false, a, /*neg_b=*/false, b,
        /*c_mod=*/(short)0, c, /*reuse_a=*/false, /*reuse_b=*/false);
  }
  const float bb = bias[n0 + lm];
#pragma unroll
  for (int i = 0; i < 8; ++i)
    h[(long)(m0 + i + 8 * half) * K + n0 + lm] = c[i] + bb;
}

// ---------------------------------------------------------------------------
// Kernel 4: per-batch selection + sparse gather-sum.
// One workgroup (256 threads) per batch row. adj rows bu,bv staged in LDS;
// selected edges (~3 per row) compacted per 256-edge chunk, then each thread
// accumulates its own output column k = threadIdx.x.
// ---------------------------------------------------------------------------
__global__ void __launch_bounds__(256)
select_accum_kernel(const float* __restrict__ adj,
                    const int* __restrict__ eu, const int* __restrict__ ev,
                    const int* __restrict__ bmin, const int* __restrict__ bmax,
                    const float* __restrict__ h, float* __restrict__ out) {
  __shared__ float rowU[N];
  __shared__ float rowV[N];
  __shared__ int   list[256];
  __shared__ int   cnt;

  const int t = threadIdx.x;
  const int b = blockIdx.x;
  const int bu = bmin[b];
  const int bv = bmax[b];

  for (int j = t; j < N; j += 256) {
    rowU[j] = adj[(long)bu * N + j];
    rowV[j] = adj[(long)bv * N + j];
  }
  if (t == 0) cnt = 0;
  __syncthreads();

  float acc = 0.0f;
  int total = 0;
  for (int c0 = 0; c0 < E; c0 += 256) {
    const int e  = c0 + t;
    const int ue = eu[e];
    const int ve = ev[e];
    // batch_edge[b,e] = adj[bu,ue]+adj[bv,ue]+adj[bu,ve]+adj[bv,ve]; 0/1 sums are exact
    const float s = rowU[ue] + rowV[ue] + rowU[ve] + rowV[ve];
    if (s == 2.0f) {
      int p = atomicAdd(&cnt, 1);  // chunk size 256 == capacity: cannot overflow
      list[p] = e;
    }
    __syncthreads();
    const int m = cnt;
    for (int i = 0; i < m; ++i)
      acc += h[(long)list[i] * K + t];
    total += m;
    __syncthreads();
    if (t == 0) cnt = 0;
    __syncthreads();
  }

  const float inv = (total > 0) ? 1.0f / (float)total : 0.0f;
  out[(long)b * K + t] = acc * inv;
}

// ---------------------------------------------------------------------------
// Host entry
// Inputs (setup_inputs order): edge_nodes[B,N], adj[N,N], inc[N,E],
//                              edge_feats[E,D], weight[D,K], bias[K]
// Output: [B,K] f32.  Workspace: h_edges (33.5 MB) + endpoint arrays.
// ---------------------------------------------------------------------------
extern "C" void kernel_launch(void* const* d_in, const int* in_sizes, int n_in,
                              void* d_out, int out_size, void* d_ws, size_t ws_size,
                              hipStream_t stream) {
  (void)in_sizes; (void)n_in; (void)out_size; (void)ws_size;
  const float* edge_nodes = (const float*)d_in[0];
  const float* adj        = (const float*)d_in[1];
  const float* inc        = (const float*)d_in[2];
  const float* edge_feats = (const float*)d_in[3];
  const float* weight     = (const float*)d_in[4];
  const float* bias       = (const float*)d_in[5];
  float* out = (float*)d_out;

  float* h  = (float*)d_ws;                        // E*K floats = 33.5 MB
  int* eu   = (int*)(h + (size_t)E * K);
  int* ev   = eu + E;
  int* bmin = ev + E;
  int* bmax = bmin + B;

  init_ws_kernel<<<E / 256, 256, 0, stream>>>(eu, ev, bmin, bmax);
  batch_endpoints_kernel<<<(B * N) / 256, 256, 0, stream>>>(edge_nodes, bmin, bmax);
  edge_endpoints_kernel<<<16384, 256, 0, stream>>>((const v4f*)inc, eu, ev);
  gemm_hedges_kernel<<<(E / 16) * (K / 16) / 8, 256, 0, stream>>>(edge_feats, weight, bias, h);
  select_accum_kernel<<<B, 256, 0, stream>>>(adj, eu, ev, bmin, bmax, h, out);
}